// QuantumLayer_74500502716746
// MI455X (gfx1250) — compile-verified
//
#include <hip/hip_runtime.h>
#include <math.h>

// ---------------------------------------------------------------------------
// Problem constants
// ---------------------------------------------------------------------------
#define BATCH     4096
#define IN_SIZE   512
#define OUT_SIZE  256
#define NQ        12
#define DEPTH     3
#define NSTATES   4096          // 2^NQ
#define BN_EPS    1e-5f

typedef float v2f __attribute__((ext_vector_type(2)));
typedef float v8f __attribute__((ext_vector_type(8)));

// ---------------------------------------------------------------------------
// Workspace layout (float offsets)
// ---------------------------------------------------------------------------
#define OFF_ENC     ((size_t)0)                       // 4096*12
#define OFF_PROBS   (OFF_ENC + (size_t)BATCH*NQ)      // 4096*12
#define OFF_OUTRAW  (OFF_PROBS + (size_t)BATCH*NQ)    // 4096*256
#define OFF_PSUM    (OFF_OUTRAW + (size_t)BATCH*OUT_SIZE)   // 64*256
#define OFF_PSUMQ   (OFF_PSUM + (size_t)64*OUT_SIZE)        // 64*256
#define OFF_SCALE2  (OFF_PSUMQ + (size_t)64*OUT_SIZE)       // 256
#define OFF_SHIFT2  (OFF_SCALE2 + (size_t)OUT_SIZE)         // 256
#define OFF_SCALE1  (OFF_SHIFT2 + (size_t)OUT_SIZE)         // 12
#define OFF_SHIFT1  (OFF_SCALE1 + (size_t)NQ)               // 12
#define OFF_U       (OFF_SHIFT1 + (size_t)NQ)               // 36*8
#define OFF_CRY     (OFF_U + (size_t)DEPTH*NQ*8)            // 33*2

// ---------------------------------------------------------------------------
// Kernel 1: precompute fused gate matrices U = Rz @ Ry @ Rx (complex 2x2)
// and CRY (cos,sin) pairs.
// ---------------------------------------------------------------------------
__global__ void gate_precompute_kernel(const float* __restrict__ rot,
                                       const float* __restrict__ ent,
                                       float* __restrict__ Umats,
                                       float* __restrict__ cryp)
{
    int tid = threadIdx.x;
    if (tid < DEPTH * NQ) {
        float a = rot[tid * 3 + 0] * 0.5f;
        float b = rot[tid * 3 + 1] * 0.5f;
        float g = rot[tid * 3 + 2] * 0.5f;
        float sa, ca, sb, cb, sg, cg;
        sincosf(a, &sa, &ca);
        sincosf(b, &sb, &cb);
        sincosf(g, &sg, &cg);
        // M1 = Ry @ Rx
        float m00r =  cb * ca, m00i =  sb * sa;
        float m01r = -sb * ca, m01i = -cb * sa;
        float m10r =  sb * ca, m10i = -cb * sa;
        float m11r =  cb * ca, m11i = -sb * sa;
        float* U = Umats + tid * 8;
        // row 0 *= exp(-i*g) = (cg, -sg)
        U[0] = m00r * cg + m00i * sg;  U[1] = m00i * cg - m00r * sg;
        U[2] = m01r * cg + m01i * sg;  U[3] = m01i * cg - m01r * sg;
        // row 1 *= exp(+i*g) = (cg, +sg)
        U[4] = m10r * cg - m10i * sg;  U[5] = m10i * cg + m10r * sg;
        U[6] = m11r * cg - m11i * sg;  U[7] = m11i * cg + m11r * sg;
    } else if (tid >= 64 && tid < 64 + DEPTH * (NQ - 1)) {
        int i = tid - 64;
        float t = ent[i] * 0.5f;
        float st, ct;
        sincosf(t, &st, &ct);
        cryp[2 * i]     = ct;
        cryp[2 * i + 1] = st;
    }
}

// ---------------------------------------------------------------------------
// Kernel 2: encoder GEMM  enc[4096x12] = x[4096x512] @ enc_w^T + enc_b
// One wave per 16-row tile; fp32 WMMA 16x16x4, N padded 12 -> 16.
// ---------------------------------------------------------------------------
__global__ void enc_gemm_kernel(const float* __restrict__ x,
                                const float* __restrict__ enc_w,
                                const float* __restrict__ enc_b,
                                float* __restrict__ enc_out)
{
    const int lane = threadIdx.x;         // 0..31 (wave32)
    const int half = lane >> 4;           // 0: K 0..1, 1: K 2..3
    const int m    = lane & 15;
    const int row  = blockIdx.x * 16 + m; // A row (M)
    // B column = m ; pad columns >= 12 with zeros (branch-free to keep EXEC full)
    const int nc   = (m < NQ) ? m : 0;
    const float bs = (m < NQ) ? 1.0f : 0.0f;

    v8f acc = {};
    const float* xr = x + (size_t)row * IN_SIZE;
    const float* wr = enc_w + (size_t)nc * IN_SIZE;
    for (int k0 = 0; k0 < IN_SIZE; k0 += 4) {
        int kk = k0 + 2 * half;
        v2f a;  a.x  = xr[kk];      a.y  = xr[kk + 1];
        v2f bv; bv.x = wr[kk] * bs; bv.y = wr[kk + 1] * bs;
        acc = __builtin_amdgcn_wmma_f32_16x16x4_f32(
                  false, a, false, bv, (short)0, acc, false, false);
    }
    if (m < NQ) {
        float bias = enc_b[m];
        #pragma unroll
        for (int r = 0; r < 8; ++r) {
            int orow = blockIdx.x * 16 + r + 8 * half;  // C/D layout: lanes16-31 -> M+8
            enc_out[(size_t)orow * NQ + m] = acc[r] + bias;
        }
    }
}

// ---------------------------------------------------------------------------
// Kernel 3: BN1 stats -> per-channel scale/shift  (12 channels, batch 4096)
// ---------------------------------------------------------------------------
__global__ void bn1_stats_kernel(const float* __restrict__ enc,
                                 const float* __restrict__ g,
                                 const float* __restrict__ bt,
                                 float* __restrict__ scale1,
                                 float* __restrict__ shift1)
{
    __shared__ float rs[256], rq[256];
    const int c = blockIdx.x, tid = threadIdx.x;
    float s = 0.f, q = 0.f;
    for (int i = tid; i < BATCH; i += 256) {
        float v = enc[(size_t)i * NQ + c];
        s += v;  q += v * v;
    }
    rs[tid] = s; rq[tid] = q;
    __syncthreads();
    for (int off = 128; off >= 1; off >>= 1) {
        if (tid < off) { rs[tid] += rs[tid + off]; rq[tid] += rq[tid + off]; }
        __syncthreads();
    }
    if (tid == 0) {
        float mean = rs[0] * (1.0f / BATCH);
        float var  = rq[0] * (1.0f / BATCH) - mean * mean;
        float is   = rsqrtf(var + BN_EPS);
        scale1[c] = g[c] * is;
        shift1[c] = bt[c] - g[c] * is * mean;
    }
}

// ---------------------------------------------------------------------------
// Kernel 4: 12-qubit statevector simulation, one workgroup per sample.
// State (4096 complex) lives in LDS; 69 gates; probs reduction at the end.
// ---------------------------------------------------------------------------
__global__ void __launch_bounds__(256)
qsim_kernel(const float* __restrict__ enc,
            const float* __restrict__ scale1,
            const float* __restrict__ shift1,
            const float* __restrict__ Umats,
            const float* __restrict__ cryp,
            float* __restrict__ probs)
{
    __shared__ float sre[NSTATES];
    __shared__ float sim_[NSTATES];
    __shared__ float qv[2 * NQ];
    __shared__ float escr[NQ];
    __shared__ float paccw[8][NQ];

    const int tid = threadIdx.x;
    const int b   = blockIdx.x;

    // BN1 + tanh for this sample's 12 channels
    if (tid < NQ)
        escr[tid] = tanhf(fmaf(enc[(size_t)b * NQ + tid], scale1[tid], shift1[tid]));
    __syncthreads();

    // amplitude encoding (serial, 12 values)
    if (tid == 0) {
        float n2 = 0.f;
        for (int q = 0; q < NQ; ++q) n2 += escr[q] * escr[q];
        float nrm = sqrtf(n2);
        float inv = (nrm > 0.f) ? 1.0f / fmaxf(nrm, 1e-30f) : 1.0f;
        for (int q = 0; q < NQ; ++q) {
            float a   = escr[q] * inv;
            float amp = fminf(fabsf(a), 1.0f);
            float ch  = sqrtf(fmaxf(1.0f - amp * amp, 0.0f));
            qv[2 * q]     = ch;
            qv[2 * q + 1] = (a < 0.f) ? -amp : amp;
        }
    }
    __syncthreads();

    // initial product state (real); qubit q -> bit (11-q)
    #pragma unroll
    for (int k = 0; k < 16; ++k) {
        int idx = tid + k * 256;
        float v = 1.0f;
        #pragma unroll
        for (int q = 0; q < NQ; ++q)
            v *= qv[2 * q + ((idx >> (11 - q)) & 1)];
        sre[idx] = v;  sim_[idx] = 0.f;
    }
    __syncthreads();

    for (int l = 0; l < DEPTH; ++l) {
        // fused Rz*Ry*Rx per qubit
        for (int q = 0; q < NQ; ++q) {
            const float* U = Umats + (l * NQ + q) * 8;
            float u00r = U[0], u00i = U[1], u01r = U[2], u01i = U[3];
            float u10r = U[4], u10i = U[5], u11r = U[6], u11i = U[7];
            int bp   = 11 - q;
            int mask = (1 << bp) - 1;
            #pragma unroll
            for (int pp = 0; pp < 8; ++pp) {
                int p  = tid + pp * 256;
                int i0 = ((p & ~mask) << 1) | (p & mask);
                int i1 = i0 | (1 << bp);
                float s0r = sre[i0], s0i = sim_[i0];
                float s1r = sre[i1], s1i = sim_[i1];
                sre[i0]  = u00r * s0r - u00i * s0i + u01r * s1r - u01i * s1i;
                sim_[i0] = u00r * s0i + u00i * s0r + u01r * s1i + u01i * s1r;
                sre[i1]  = u10r * s0r - u10i * s0i + u11r * s1r - u11i * s1i;
                sim_[i1] = u10r * s0i + u10i * s0r + u11r * s1i + u11i * s1r;
            }
            __syncthreads();
        }
        // CRY(q -> q+1): apply real Ry to target where control bit == 1
        for (int q = 0; q < NQ - 1; ++q) {
            float c = cryp[(l * (NQ - 1) + q) * 2];
            float s = cryp[(l * (NQ - 1) + q) * 2 + 1];
            int bt    = 10 - q;            // target bit; control bit = bt+1
            int maskt = (1 << bt) - 1;
            #pragma unroll
            for (int pp = 0; pp < 4; ++pp) {
                int p  = tid + pp * 256;
                int i0 = ((p & ~maskt) << 2) | (1 << (bt + 1)) | (p & maskt);
                int i1 = i0 | (1 << bt);
                float s0r = sre[i0], s0i = sim_[i0];
                float s1r = sre[i1], s1i = sim_[i1];
                sre[i0]  = c * s0r - s * s1r;   sim_[i0] = c * s0i - s * s1i;
                sre[i1]  = s * s0r + c * s1r;   sim_[i1] = s * s0i + c * s1i;
            }
            __syncthreads();
        }
    }

    // probs[q] = sum_idx |s|^2 * bit(idx, q)
    float pq[NQ];
    #pragma unroll
    for (int q = 0; q < NQ; ++q) pq[q] = 0.f;
    #pragma unroll
    for (int k = 0; k < 16; ++k) {
        int idx = tid + k * 256;
        float r = sre[idx], im = sim_[idx];
        float pw = r * r + im * im;
        #pragma unroll
        for (int q = 0; q < NQ; ++q)
            pq[q] = fmaf((float)((idx >> (11 - q)) & 1), pw, pq[q]);
    }
    #pragma unroll
    for (int q = 0; q < NQ; ++q)
        for (int off = 16; off >= 1; off >>= 1)
            pq[q] += __shfl_xor(pq[q], off, 32);
    int wv = tid >> 5, ln = tid & 31;
    if (ln == 0) {
        #pragma unroll
        for (int q = 0; q < NQ; ++q) paccw[wv][q] = pq[q];
    }
    __syncthreads();
    if (tid < NQ) {
        float sum = 0.f;
        #pragma unroll
        for (int w = 0; w < 8; ++w) sum += paccw[w][tid];
        probs[(size_t)b * NQ + tid] = sum;
    }
}

// ---------------------------------------------------------------------------
// Kernel 5: decoder GEMM  outraw[4096x256] = probs[4096x12] @ dec_w^T + dec_b
// K = 12 = exactly three 16x16x4 WMMA steps; one wave per 16x16 tile.
// ---------------------------------------------------------------------------
__global__ void dec_gemm_kernel(const float* __restrict__ probs,
                                const float* __restrict__ dec_w,
                                const float* __restrict__ dec_b,
                                float* __restrict__ outraw)
{
    const int lane  = threadIdx.x;
    const int half  = lane >> 4;
    const int m     = lane & 15;
    const int mtile = blockIdx.x;
    const int ntile = blockIdx.y;
    const int row   = mtile * 16 + m;
    const int n     = ntile * 16 + m;

    v8f acc = {};
    const float* pr = probs + (size_t)row * NQ;
    const float* wr = dec_w + (size_t)n * NQ;
    #pragma unroll
    for (int k0 = 0; k0 < NQ; k0 += 4) {
        int kk = k0 + 2 * half;
        v2f a;  a.x  = pr[kk]; a.y  = pr[kk + 1];
        v2f bv; bv.x = wr[kk]; bv.y = wr[kk + 1];
        acc = __builtin_amdgcn_wmma_f32_16x16x4_f32(
                  false, a, false, bv, (short)0, acc, false, false);
    }
    float bias = dec_b[n];
    #pragma unroll
    for (int r = 0; r < 8; ++r) {
        int orow = mtile * 16 + r + 8 * half;
        outraw[(size_t)orow * OUT_SIZE + n] = acc[r] + bias;
    }
}

// ---------------------------------------------------------------------------
// Kernel 6/7/8: BN2 — deterministic two-stage reduction, then apply.
// ---------------------------------------------------------------------------
__global__ void bn2_partial_kernel(const float* __restrict__ outraw,
                                   float* __restrict__ psum,
                                   float* __restrict__ psumq)
{
    const int c = threadIdx.x, blk = blockIdx.x;
    float s = 0.f, q = 0.f;
    const int base = blk * 64;
    for (int r = 0; r < 64; ++r) {
        float v = outraw[(size_t)(base + r) * OUT_SIZE + c];
        s += v;  q += v * v;
    }
    psum[blk * OUT_SIZE + c]  = s;
    psumq[blk * OUT_SIZE + c] = q;
}

__global__ void bn2_finalize_kernel(const float* __restrict__ psum,
                                    const float* __restrict__ psumq,
                                    const float* __restrict__ g,
                                    const float* __restrict__ bt,
                                    float* __restrict__ scale2,
                                    float* __restrict__ shift2)
{
    const int c = threadIdx.x;
    float s = 0.f, q = 0.f;
    for (int b2 = 0; b2 < 64; ++b2) {
        s += psum[b2 * OUT_SIZE + c];
        q += psumq[b2 * OUT_SIZE + c];
    }
    float mean = s * (1.0f / BATCH);
    float var  = q * (1.0f / BATCH) - mean * mean;
    float is   = rsqrtf(var + BN_EPS);
    scale2[c] = g[c] * is;
    shift2[c] = bt[c] - g[c] * is * mean;
}

__global__ void bn2_apply_kernel(const float* __restrict__ outraw,
                                 const float* __restrict__ scale2,
                                 const float* __restrict__ shift2,
                                 float* __restrict__ out)
{
    const int c = threadIdx.x, i = blockIdx.x;
    size_t idx = (size_t)i * OUT_SIZE + c;
    out[idx] = fmaf(outraw[idx], scale2[c], shift2[c]);
}

// ---------------------------------------------------------------------------
// Launch
// ---------------------------------------------------------------------------
extern "C" void kernel_launch(void* const* d_in, const int* in_sizes, int n_in,
                              void* d_out, int out_size, void* d_ws, size_t ws_size,
                              hipStream_t stream)
{
    (void)in_sizes; (void)n_in; (void)out_size; (void)ws_size;
    const float* x     = (const float*)d_in[0];
    const float* enc_w = (const float*)d_in[1];
    const float* enc_b = (const float*)d_in[2];
    const float* rot   = (const float*)d_in[3];
    const float* ent   = (const float*)d_in[4];
    const float* dec_w = (const float*)d_in[5];
    const float* dec_b = (const float*)d_in[6];
    const float* bn1g  = (const float*)d_in[7];
    const float* bn1b  = (const float*)d_in[8];
    const float* bn2g  = (const float*)d_in[9];
    const float* bn2b  = (const float*)d_in[10];

    float* ws     = (float*)d_ws;
    float* enc    = ws + OFF_ENC;
    float* probs  = ws + OFF_PROBS;
    float* outraw = ws + OFF_OUTRAW;
    float* psum   = ws + OFF_PSUM;
    float* psumq  = ws + OFF_PSUMQ;
    float* scale2 = ws + OFF_SCALE2;
    float* shift2 = ws + OFF_SHIFT2;
    float* scale1 = ws + OFF_SCALE1;
    float* shift1 = ws + OFF_SHIFT1;
    float* Umats  = ws + OFF_U;
    float* cryp   = ws + OFF_CRY;

    gate_precompute_kernel<<<1, 128, 0, stream>>>(rot, ent, Umats, cryp);
    enc_gemm_kernel<<<BATCH / 16, 32, 0, stream>>>(x, enc_w, enc_b, enc);
    bn1_stats_kernel<<<NQ, 256, 0, stream>>>(enc, bn1g, bn1b, scale1, shift1);
    qsim_kernel<<<BATCH, 256, 0, stream>>>(enc, scale1, shift1, Umats, cryp, probs);
    dim3 dgrid(BATCH / 16, OUT_SIZE / 16);
    dec_gemm_kernel<<<dgrid, 32, 0, stream>>>(probs, dec_w, dec_b, outraw);
    bn2_partial_kernel<<<64, 256, 0, stream>>>(outraw, psum, psumq);
    bn2_finalize_kernel<<<1, 256, 0, stream>>>(psum, psumq, bn2g, bn2b, scale2, shift2);
    bn2_apply_kernel<<<BATCH, 256, 0, stream>>>(outraw, scale2, shift2, (float*)d_out);
}